// GraphNet_45389214384568
// MI455X (gfx1250) — compile-verified
//
#include <hip/hip_runtime.h>
#include <math.h>

#define NN 100000
#define NE 1600000
#define C 128
#define OUTC 10
#define LDS_PITCH 132  // 128 + 4 pad: avoids 64-bank conflicts on row-major column reads

typedef __attribute__((ext_vector_type(2))) float v2f;
typedef __attribute__((ext_vector_type(8))) float v8f;

// ---------------------------------------------------------------- zero fill
__global__ void fill_zero_f4(float4* __restrict__ p, int n4) {
  int i = blockIdx.x * blockDim.x + threadIdx.x;
  if (i < n4) { float4 z = {0.f, 0.f, 0.f, 0.f}; p[i] = z; }
}

// ------------------------------------------------- edge scatter (message+agg)
// One wave per edge; lane handles 4 channels (float4 gather, 4 f32 atomics).
// x[src] (51.2MB) and agg (51.2MB) both fit in the 192MB L2 -> L2-rate atomics.
__global__ void scatter_edges(const float* __restrict__ x,
                              const int* __restrict__ src,
                              const int* __restrict__ dst,
                              const float* __restrict__ ew,
                              float* __restrict__ agg) {
  unsigned gid = blockIdx.x * blockDim.x + threadIdx.x;
  unsigned edge = gid >> 5;
  unsigned lane = gid & 31u;
  if (edge >= NE) return;
  int s = src[edge];
  int d = dst[edge];
  float w = ew[edge];
  float4 v = ((const float4*)(x + (size_t)s * C))[lane];
  float* ap = agg + (size_t)d * C + lane * 4;
  atomicAdd(ap + 0, v.x * w);
  atomicAdd(ap + 1, v.y * w);
  atomicAdd(ap + 2, v.z * w);
  atomicAdd(ap + 3, v.w * w);
}

// ------------------------------------- dual GEMM + bias + relu via f32 WMMA
// out[16 nodes, 128] = relu(agg@Wrel + x@Wroot + b). Block = 256 thr = 8 waves;
// wave w owns output columns [16w,16w+16). A tiles staged in LDS (8x reuse).
__global__ void __launch_bounds__(256)
gconv_update(const float* __restrict__ agg,
             const float* __restrict__ x,
             const float* __restrict__ Wrel,   // [128,128] row-major K x N
             const float* __restrict__ Wroot,  // [128,128]
             const float* __restrict__ bias,   // [128]
             float* __restrict__ out) {
  __shared__ float sAgg[16 * LDS_PITCH];
  __shared__ float sX[16 * LDS_PITCH];

  const int tid = threadIdx.x;
  const int row0 = blockIdx.x * 16;  // 100000 / 16 = 6250 exact, no tail

  // Stage both 16x128 A tiles, coalesced float4 (512 float4 per tile).
  {
    const float4* ga = (const float4*)(agg + (size_t)row0 * C);
    const float4* gx = (const float4*)(x + (size_t)row0 * C);
    for (int i = tid; i < 512; i += 256) {
      int r = i >> 5;        // 32 float4 per row
      int c4 = i & 31;
      float4 va = ga[i];
      float4 vx = gx[i];
      float* da = &sAgg[r * LDS_PITCH + c4 * 4];
      da[0] = va.x; da[1] = va.y; da[2] = va.z; da[3] = va.w;
      float* dx = &sX[r * LDS_PITCH + c4 * 4];
      dx[0] = vx.x; dx[1] = vx.y; dx[2] = vx.z; dx[3] = vx.w;
    }
  }
  __syncthreads();

  const int lane = tid & 31;
  const int wave = tid >> 5;       // column tile 0..7
  const int n0 = wave * 16;
  const int half = lane >> 4;      // 0: K+0/1, M=r ; 1: K+2/3, M=r+8
  const int l = lane & 15;
  const int kofs = 2 * half;

  const float* aRow = &sAgg[l * LDS_PITCH + kofs];
  const float* xRow = &sX[l * LDS_PITCH + kofs];
  const float* wr = Wrel + n0 + l;   // column n0+l of Wrel
  const float* wo = Wroot + n0 + l;

  v8f accR = {};  // agg @ Wrel
  v8f accX = {};  // x   @ Wroot   (independent chains -> better WMMA pipelining)
#pragma unroll 4
  for (int k = 0; k < C; k += 4) {
    v2f a0; a0.x = aRow[k];               a0.y = aRow[k + 1];
    v2f b0; b0.x = wr[(k + kofs) * C];    b0.y = wr[(k + kofs + 1) * C];
    accR = __builtin_amdgcn_wmma_f32_16x16x4_f32(false, a0, false, b0,
                                                 (short)0, accR, false, false);
    v2f a1; a1.x = xRow[k];               a1.y = xRow[k + 1];
    v2f b1; b1.x = wo[(k + kofs) * C];    b1.y = wo[(k + kofs + 1) * C];
    accX = __builtin_amdgcn_wmma_f32_16x16x4_f32(false, a1, false, b1,
                                                 (short)0, accX, false, false);
  }

  // D layout: VGPR r -> row (row0 + r + 8*half), col (n0 + l)
  float bv = bias[n0 + l];
  float* orow = out + (size_t)(row0 + half * 8) * C + n0 + l;
#pragma unroll
  for (int r = 0; r < 8; r++) {
    float v = accR[r] + accX[r] + bv;
    v = fmaxf(v, 0.f);
    orow[(size_t)r * C] = v;
  }
}

// --------------------------------------- head: x3 @ W_lin + b, log_softmax
// One wave per node; lane covers 4 of 128 channels; wave-reduce per class.
__global__ void head_kernel(const float* __restrict__ x3,
                            const float* __restrict__ Wlin,  // [128,10]
                            const float* __restrict__ blin,  // [10]
                            float* __restrict__ out) {
  unsigned gid = blockIdx.x * blockDim.x + threadIdx.x;
  unsigned node = gid >> 5;
  unsigned lane = gid & 31u;
  if (node >= NN) return;
  const float* xr = x3 + (size_t)node * C;
  float xv[4];
#pragma unroll
  for (int j = 0; j < 4; j++) xv[j] = xr[lane + 32 * j];

  float logits[OUTC];
#pragma unroll
  for (int c = 0; c < OUTC; c++) {
    float p = 0.f;
#pragma unroll
    for (int j = 0; j < 4; j++) p += xv[j] * Wlin[(lane + 32 * j) * OUTC + c];
    for (int off = 16; off > 0; off >>= 1) p += __shfl_down(p, off, 32);
    logits[c] = p;  // valid in lane 0
  }
  if (lane == 0) {
    float m = -INFINITY;
#pragma unroll
    for (int c = 0; c < OUTC; c++) { logits[c] += blin[c]; m = fmaxf(m, logits[c]); }
    float s = 0.f;
#pragma unroll
    for (int c = 0; c < OUTC; c++) s += __expf(logits[c] - m);
    float lse = m + __logf(s);
    float* orow = out + (size_t)node * OUTC;
#pragma unroll
    for (int c = 0; c < OUTC; c++) orow[c] = logits[c] - lse;
  }
}

// ---------------------------------------------------------------- launcher
extern "C" void kernel_launch(void* const* d_in, const int* in_sizes, int n_in,
                              void* d_out, int out_size, void* d_ws, size_t ws_size,
                              hipStream_t stream) {
  const float* x0  = (const float*)d_in[0];
  const int*   ei  = (const int*)d_in[1];
  const int*   src = ei;
  const int*   dst = ei + NE;
  const float* ew  = (const float*)d_in[2];
  const float* W1r = (const float*)d_in[3];
  const float* b1  = (const float*)d_in[4];
  const float* W1o = (const float*)d_in[5];
  const float* W2r = (const float*)d_in[6];
  const float* b2  = (const float*)d_in[7];
  const float* W2o = (const float*)d_in[8];
  const float* W3r = (const float*)d_in[9];
  const float* b3  = (const float*)d_in[10];
  const float* W3o = (const float*)d_in[11];
  const float* Wl  = (const float*)d_in[12];
  const float* bl  = (const float*)d_in[13];
  float* outp = (float*)d_out;

  // workspace: agg | bufA | bufB, each 100000*128 f32 (51.2MB) -> 153.6MB total
  float* agg  = (float*)d_ws;
  float* bufA = agg + (size_t)NN * C;
  float* bufB = bufA + (size_t)NN * C;

  const int n4 = NN * C / 4;                 // 3,200,000 float4
  const unsigned zgrid = (n4 + 255) / 256;   // 12,500
  const unsigned sgrid = (unsigned)(((size_t)NE * 32 + 255) / 256);  // 200,000
  const unsigned ggrid = NN / 16;            // 6,250 (exact)
  const unsigned hgrid = (unsigned)(((size_t)NN * 32 + 255) / 256); // 12,500

  // layer 1: x0 -> bufA
  fill_zero_f4<<<zgrid, 256, 0, stream>>>((float4*)agg, n4);
  scatter_edges<<<sgrid, 256, 0, stream>>>(x0, src, dst, ew, agg);
  gconv_update<<<ggrid, 256, 0, stream>>>(agg, x0, W1r, W1o, b1, bufA);

  // layer 2: bufA -> bufB
  fill_zero_f4<<<zgrid, 256, 0, stream>>>((float4*)agg, n4);
  scatter_edges<<<sgrid, 256, 0, stream>>>(bufA, src, dst, ew, agg);
  gconv_update<<<ggrid, 256, 0, stream>>>(agg, bufA, W2r, W2o, b2, bufB);

  // layer 3: bufB -> bufA
  fill_zero_f4<<<zgrid, 256, 0, stream>>>((float4*)agg, n4);
  scatter_edges<<<sgrid, 256, 0, stream>>>(bufB, src, dst, ew, agg);
  gconv_update<<<ggrid, 256, 0, stream>>>(agg, bufB, W3r, W3o, b3, bufA);

  // head
  head_kernel<<<hgrid, 256, 0, stream>>>(bufA, Wl, bl, outp);
}